// AAMsoftmaxloss_8589934592489
// MI455X (gfx1250) — compile-verified
//
#include <hip/hip_runtime.h>
#include <math.h>

// ---------------------------------------------------------------------------
// ArcFace (AAM-softmax) loss, fused, fp32 WMMA path for gfx1250 (MI455X).
// GEMM is compute/L2-bound: block covers 64 rows so each B (weight) fragment
// feeds 4 WMMAs, cutting weight L2 traffic 4x vs a 16-row block.
// ---------------------------------------------------------------------------

typedef __attribute__((ext_vector_type(2))) float v2f;
typedef __attribute__((ext_vector_type(8))) float v8f;

#define D_DIM 256
#define B_DIM 1024
#define C_DIM 100000
#define NT_COL (C_DIM / 16)          // 6250 column tiles of 16
#define WAVES_PER_BLOCK 8
#define TILES_PER_WAVE 16
#define TILES_PER_BLOCK (WAVES_PER_BLOCK * TILES_PER_WAVE)   // 128 col tiles
#define NUM_COL_BLOCKS ((NT_COL + TILES_PER_BLOCK - 1) / TILES_PER_BLOCK) // 49
#define M_TILES 4                    // row tiles of 16 per block
#define M_BLOCK (M_TILES * 16)       // 64 rows per block
#define LDS_STRIDE 260               // padded A-tile stride: conflict-free b64 reads
                                     // (16*260 % 64 == 0 -> same bank map per row tile)

// ArcFace constants (margin = 0.1, scale = 50)
#define AAM_COS_M 0.99500416527802576610f
#define AAM_SIN_M 0.09983341664682815230f
#define AAM_TH   (-0.99500416527802576610f)   // cos(pi - m)
#define AAM_MM    0.00998334166468281523f     // sin(pi - m) * m
#define AAM_S     50.0f

__device__ __forceinline__ float waveReduceSum32(float v) {
  v += __shfl_xor(v, 16, 32);
  v += __shfl_xor(v, 8, 32);
  v += __shfl_xor(v, 4, 32);
  v += __shfl_xor(v, 2, 32);
  v += __shfl_xor(v, 1, 32);
  return v;
}

// ---------------------------------------------------------------------------
// Kernel 1: per-row inverse L2 norm (wave per row, 256 floats -> 8/lane)
// ---------------------------------------------------------------------------
__global__ __launch_bounds__(256)
void row_inv_norm_kernel(const float* __restrict__ src, float* __restrict__ inv,
                         int rows) {
  const int wave = threadIdx.x >> 5;
  const int lane = threadIdx.x & 31;
  const int row = blockIdx.x * WAVES_PER_BLOCK + wave;
  if (row >= rows) return;
  const float4* p = (const float4*)(src + (size_t)row * D_DIM) + lane * 2;
  float4 a = p[0];
  float4 b = p[1];
  float s = a.x * a.x + a.y * a.y + a.z * a.z + a.w * a.w
          + b.x * b.x + b.y * b.y + b.z * b.z + b.w * b.w;
  s = waveReduceSum32(s);
  if (lane == 0) {
    inv[row] = 1.0f / fmaxf(sqrtf(s), 1e-12f);
  }
}

// ---------------------------------------------------------------------------
// Kernel 2: target cosine per sample: cos_t[b] = <x_b, w_label[b]> * invx * invw
// ---------------------------------------------------------------------------
__global__ __launch_bounds__(256)
void target_cos_kernel(const float* __restrict__ x, const float* __restrict__ w,
                       const int* __restrict__ label,
                       const float* __restrict__ xinv, const float* __restrict__ winv,
                       float* __restrict__ cost) {
  const int wave = threadIdx.x >> 5;
  const int lane = threadIdx.x & 31;
  const int b = blockIdx.x * WAVES_PER_BLOCK + wave;
  if (b >= B_DIM) return;
  const int lbl = label[b];
  const float4* xp = (const float4*)(x + (size_t)b * D_DIM) + lane * 2;
  const float4* wp = (const float4*)(w + (size_t)lbl * D_DIM) + lane * 2;
  float4 xa = xp[0], xb = xp[1];
  float4 wa = wp[0], wb = wp[1];
  float s = xa.x * wa.x + xa.y * wa.y + xa.z * wa.z + xa.w * wa.w
          + xb.x * wb.x + xb.y * wb.y + xb.z * wb.z + xb.w * wb.w;
  s = waveReduceSum32(s);
  if (lane == 0) cost[b] = s * xinv[b] * winv[lbl];
}

// ---------------------------------------------------------------------------
// Kernel 3: fused GEMM (x . w^T) + scale-to-cosine + exp(50*cos - 50) row sums.
// Grid: (B/64, NUM_COL_BLOCKS). Block: 256 threads (8 waves), 64-row A tile
// in LDS. Per K-step each wave loads ONE B fragment (global b64) and runs 4
// WMMAs against 4 LDS A fragments -> 4 independent accumulator chains and
// 4x less weight traffic per FLOP. Deterministic: partial[colblock][row].
// ---------------------------------------------------------------------------
__global__ __launch_bounds__(256)
void aam_gemm_sumexp_kernel(const float* __restrict__ x, const float* __restrict__ w,
                            const float* __restrict__ xinv, const float* __restrict__ winv,
                            float* __restrict__ partial) {
  __shared__ float As[M_BLOCK * LDS_STRIDE];     // 64x256 padded  (~66.5 KB)
  __shared__ float sxinv[M_BLOCK];
  __shared__ float wsums[WAVES_PER_BLOCK][M_BLOCK];

  const int tid = threadIdx.x;
  const int rowBase = blockIdx.x * M_BLOCK;

  // Stage the 64x256 A tile into LDS (padded stride; float4 coalesced loads).
  {
    const float4* xsrc = (const float4*)(x + (size_t)rowBase * D_DIM);
    #pragma unroll
    for (int i = 0; i < 16; ++i) {
      int idx = tid + i * 256;        // 4096 float4 elements total
      int r = idx >> 6;               // 64 float4 per row
      int c = (idx & 63) << 2;
      float4 v = xsrc[idx];
      float* dst = &As[r * LDS_STRIDE + c];
      dst[0] = v.x; dst[1] = v.y; dst[2] = v.z; dst[3] = v.w;
    }
  }
  if (tid < M_BLOCK) sxinv[tid] = xinv[rowBase + tid];
  __syncthreads();

  const int wave = tid >> 5;
  const int lane = tid & 31;
  const int laneN = lane & 15;   // A-frag M / B-frag N / C-frag N
  const int laneH = lane >> 4;   // half: selects K pair (A/B), M+8 (C)

  float rowacc[M_TILES][8];
  #pragma unroll
  for (int r = 0; r < M_TILES; ++r)
    #pragma unroll
    for (int i = 0; i < 8; ++i) rowacc[r][i] = 0.0f;

  const int tileBase = blockIdx.y * TILES_PER_BLOCK + wave * TILES_PER_WAVE;
  const float* AfragBase = &As[laneN * LDS_STRIDE + 2 * laneH];

  for (int j = 0; j < TILES_PER_WAVE; ++j) {
    const int t = tileBase + j;
    if (t >= NT_COL) break;                     // contiguous range per wave

    const float* wp = w + (size_t)(t * 16 + laneN) * D_DIM + 2 * laneH;

    v8f acc[M_TILES];
    #pragma unroll
    for (int r = 0; r < M_TILES; ++r)
      acc[r] = (v8f){0.f, 0.f, 0.f, 0.f, 0.f, 0.f, 0.f, 0.f};

    #pragma unroll 8
    for (int k0 = 0; k0 < D_DIM; k0 += 4) {
      v2f b = *(const v2f*)(wp + k0);                          // global_load_b64
      #pragma unroll
      for (int r = 0; r < M_TILES; ++r) {
        v2f a = *(const v2f*)(AfragBase + r * 16 * LDS_STRIDE + k0); // ds_load_b64
        acc[r] = __builtin_amdgcn_wmma_f32_16x16x4_f32(
            false, a, false, b, (short)0, acc[r], false, false);
      }
    }

    const float winv0 = winv[t * 16 + laneN];
    #pragma unroll
    for (int r = 0; r < M_TILES; ++r) {
      #pragma unroll
      for (int i = 0; i < 8; ++i) {
        const float invxr = sxinv[r * 16 + i + 8 * laneH];
        const float c0 = acc[r][i] * invxr * winv0;
        rowacc[r][i] += __expf(AAM_S * c0 - AAM_S);
      }
    }
  }

  // Reduce across the 16 lanes of each half (same output row, distinct cols).
  #pragma unroll
  for (int r = 0; r < M_TILES; ++r) {
    #pragma unroll
    for (int i = 0; i < 8; ++i) {
      float v = rowacc[r][i];
      v += __shfl_xor(v, 1, 32);
      v += __shfl_xor(v, 2, 32);
      v += __shfl_xor(v, 4, 32);
      v += __shfl_xor(v, 8, 32);
      if (laneN == 0) wsums[wave][r * 16 + i + 8 * laneH] = v;
    }
  }
  __syncthreads();

  // Deterministic per-block reduction over the 8 waves; unique output slot.
  if (tid < M_BLOCK) {
    float s = 0.0f;
    #pragma unroll
    for (int wv = 0; wv < WAVES_PER_BLOCK; ++wv) s += wsums[wv][tid];
    partial[(size_t)blockIdx.y * B_DIM + rowBase + tid] = s;
  }
}

// ---------------------------------------------------------------------------
// Kernel 4: finalize. Deterministic partial reduction per row, swap the label
// column's cosine term for its margin (phi) term, logsumexp, mean CE loss.
// ---------------------------------------------------------------------------
__global__ __launch_bounds__(256)
void final_loss_kernel(const float* __restrict__ partial,
                       const float* __restrict__ cost,
                       float* __restrict__ out) {
  __shared__ float sbuf[256];
  const int tid = threadIdx.x;
  float local = 0.0f;
  for (int b = tid; b < B_DIM; b += 256) {
    float sum = 0.0f;
    for (int g = 0; g < NUM_COL_BLOCKS; ++g) sum += partial[(size_t)g * B_DIM + b];
    const float c = cost[b];
    float s2 = 1.0f - c * c;
    s2 = fmaxf(0.0f, fminf(1.0f, s2));
    const float sn = sqrtf(s2);
    const float phi = c * AAM_COS_M - sn * AAM_SIN_M;
    const float psel = ((c - AAM_TH) > 0.0f) ? phi : (c - AAM_MM);
    const float tlogit = AAM_S * psel;
    // replace plain-cosine label term with margin term (fixed shift of 50;
    // logits are bounded above by 50, so exp(l-50) can never overflow)
    const float adj = sum - __expf(AAM_S * c - AAM_S) + __expf(tlogit - AAM_S);
    const float logz = logf(adj) + AAM_S;
    local += logz - tlogit;
  }
  sbuf[tid] = local;
  __syncthreads();
  #pragma unroll
  for (int s = 128; s > 0; s >>= 1) {
    if (tid < s) sbuf[tid] += sbuf[tid + s];
    __syncthreads();
  }
  if (tid == 0) out[0] = sbuf[0] * (1.0f / (float)B_DIM);
}

// ---------------------------------------------------------------------------
// Launch
// ---------------------------------------------------------------------------
extern "C" void kernel_launch(void* const* d_in, const int* in_sizes, int n_in,
                              void* d_out, int out_size, void* d_ws, size_t ws_size,
                              hipStream_t stream) {
  (void)in_sizes; (void)n_in; (void)out_size; (void)ws_size;
  const float* x = (const float*)d_in[0];      // [B, D] fp32
  const float* w = (const float*)d_in[1];      // [C, D] fp32
  const int* label = (const int*)d_in[2];      // [B]
  float* out = (float*)d_out;                  // scalar loss

  float* ws = (float*)d_ws;
  float* xinv = ws;                            // [B]
  float* winv = xinv + B_DIM;                  // [C]
  float* cost = winv + C_DIM;                  // [B]
  float* partial = cost + B_DIM;               // [NUM_COL_BLOCKS, B]

  row_inv_norm_kernel<<<B_DIM / WAVES_PER_BLOCK, 256, 0, stream>>>(x, xinv, B_DIM);
  row_inv_norm_kernel<<<(C_DIM + WAVES_PER_BLOCK - 1) / WAVES_PER_BLOCK, 256, 0, stream>>>(w, winv, C_DIM);
  target_cos_kernel<<<B_DIM / WAVES_PER_BLOCK, 256, 0, stream>>>(x, w, label, xinv, winv, cost);

  dim3 grid(B_DIM / M_BLOCK, NUM_COL_BLOCKS);
  aam_gemm_sumexp_kernel<<<grid, 256, 0, stream>>>(x, w, xinv, winv, partial);

  final_loss_kernel<<<1, 256, 0, stream>>>(partial, cost, out);
}